// VAE_61564061221261
// MI455X (gfx1250) — compile-verified
//
#include <hip/hip_runtime.h>

// CDNA5 / gfx1250, wave32.
typedef __attribute__((ext_vector_type(2))) float v2f;
typedef __attribute__((ext_vector_type(8))) float v8f;

#define NROWS   16384
#define DIM     64
#define ROWS_PER_WAVE 16
#define NWAVES  (NROWS / ROWS_PER_WAVE)      // 1024
#define NPART   (NWAVES * 2)                 // 2048 double partials in d_ws

// One v_wmma_f32_16x16x4_f32 row-sum step: D = A * ones(4x16) + C.
static __device__ __forceinline__ v8f rowsum_acc(v2f a, v2f ones, v8f c) {
  return __builtin_amdgcn_wmma_f32_16x16x4_f32(
      /*neg_a=*/false, a, /*neg_b=*/false, ones,
      /*c_mod=*/(short)0, c, /*reuse_a=*/false, /*reuse_b=*/false);
}

__global__ __launch_bounds__(128) void vae_kl_stage1(
    const float* __restrict__ loc_g, const float* __restrict__ loc_u,
    const float* __restrict__ scl_g, const float* __restrict__ scl_u,
    double* __restrict__ partials)
{
  const unsigned tid  = threadIdx.x;
  const unsigned lane = tid & 31u;         // wave32
  const unsigned hi   = lane >> 4;         // 0: K-slots {0,1}, 1: K-slots {2,3}
  const unsigned mlo  = lane & 15u;        // row within the 16-row tile (M)
  const unsigned wave = (blockIdx.x * blockDim.x + tid) >> 5;
  const unsigned row  = wave * ROWS_PER_WAVE + mlo;
  const unsigned base = row * DIM + 2u * hi;

  v2f ones; ones.x = 1.0f; ones.y = 1.0f;

  // 12 row-sum accumulators (gs side S*, uns side T*)
  v8f aS1 = {}, aS2 = {}, aS3 = {}, aS4 = {}, aS5 = {}, aS6 = {};
  v8f aT1 = {}, aT2 = {}, aT3 = {}, aT4 = {}, aT5 = {}, aT6 = {};

  #pragma unroll 4
  for (int c = 0; c < DIM / 4; ++c) {
    const unsigned off = base + 4u * (unsigned)c;
    v2f g  = *(const v2f*)(loc_g + off);
    v2f u  = *(const v2f*)(loc_u + off);
    v2f sg = *(const v2f*)(scl_g + off);
    v2f su = *(const v2f*)(scl_u + off);

    v2f sg2 = sg * sg;
    v2f su2 = su * su;
    v2f ig, iu;                                 // 1/(4*sigma^2)
    ig.x = 0.25f * __builtin_amdgcn_rcpf(sg2.x);
    ig.y = 0.25f * __builtin_amdgcn_rcpf(sg2.y);
    iu.x = 0.25f * __builtin_amdgcn_rcpf(su2.x);
    iu.y = 0.25f * __builtin_amdgcn_rcpf(su2.y);
    v2f g2  = g * g,   u2  = u * u;
    v2f gi  = g * ig,  ui  = u * iu;
    v2f g2i = g2 * ig, u2i = u2 * iu;

    aS1 = rowsum_acc(ig,  ones, aS1);   // sum 1/(4 sg2)
    aS2 = rowsum_acc(sg2, ones, aS2);   // sum sg2
    aS3 = rowsum_acc(g,   ones, aS3);   // sum g
    aS4 = rowsum_acc(g2,  ones, aS4);   // sum g^2
    aS5 = rowsum_acc(g2i, ones, aS5);   // sum g^2/(4 sg2)
    aS6 = rowsum_acc(gi,  ones, aS6);   // sum g/(4 sg2)

    aT1 = rowsum_acc(iu,  ones, aT1);
    aT2 = rowsum_acc(su2, ones, aT2);
    aT3 = rowsum_acc(u,   ones, aT3);
    aT4 = rowsum_acc(u2,  ones, aT4);
    aT5 = rowsum_acc(u2i, ones, aT5);
    aT6 = rowsum_acc(ui,  ones, aT6);
  }

  // Each accumulator VGPR v on this lane holds the row-sum for one of 8 rows
  // (same M for all 12 accumulators); the two lane-halves cover the other 8.
  // Summing C'(m) over v for both halves covers all 16 rows exactly once.
  double partial = 0.0;
  #pragma unroll
  for (int v = 0; v < 8; ++v) {
    float S1 = aS1[v], S2 = aS2[v], S3 = aS3[v];
    float S4 = aS4[v], S5 = aS5[v], S6 = aS6[v];
    float T1 = aT1[v], T2 = aT2[v], T3 = aT3[v];
    float T4 = aT4[v], T5 = aT5[v], T6 = aT6[v];
    float Cn = T2 * S1 + S2 * T1
             + (S1 * T4 - 2.0f * S6 * T3 + 64.0f * S5)
             + (T1 * S4 - 2.0f * T6 * S3 + 64.0f * T5);
    partial += (double)Cn;
  }

  if (mlo == 0) {                      // lanes 0 and 16: one slot per half
    partials[wave * 2u + hi] = partial;
  }
}

__global__ __launch_bounds__(256) void vae_kl_stage2(
    const double* __restrict__ partials, float* __restrict__ out)
{
  __shared__ double sm[256];
  double s = 0.0;
  for (int i = (int)threadIdx.x; i < NPART; i += 256) s += partials[i];
  sm[threadIdx.x] = s;
  __syncthreads();
  for (int w = 128; w > 0; w >>= 1) {
    if ((int)threadIdx.x < w) sm[threadIdx.x] += sm[threadIdx.x + w];
    __syncthreads();
  }
  if (threadIdx.x == 0) {
    const double scale = (double)NROWS * (double)DIM * (double)DIM;
    out[0] = (float)(sm[0] / scale - 0.5);
  }
}

extern "C" void kernel_launch(void* const* d_in, const int* in_sizes, int n_in,
                              void* d_out, int out_size, void* d_ws, size_t ws_size,
                              hipStream_t stream) {
  (void)in_sizes; (void)n_in; (void)out_size; (void)ws_size;
  const float* z_loc_gs   = (const float*)d_in[0];
  const float* z_loc_uns  = (const float*)d_in[1];
  const float* z_scale_gs = (const float*)d_in[2];
  const float* z_scale_uns= (const float*)d_in[3];
  float*  out      = (float*)d_out;
  double* partials = (double*)d_ws;    // NPART * 8 bytes = 16 KB

  const int threads = 128;                         // 4 waves/block
  const int blocks  = (NWAVES * 32) / threads;     // 256 blocks
  vae_kl_stage1<<<blocks, threads, 0, stream>>>(
      z_loc_gs, z_loc_uns, z_scale_gs, z_scale_uns, partials);
  vae_kl_stage2<<<1, 256, 0, stream>>>(partials, out);
}